// GConvGRU_Baseline_73332271612042
// MI455X (gfx1250) — compile-verified
//
#include <hip/hip_runtime.h>
#include <math.h>

// ---------------------------------------------------------------------------
// GConvGRU (ChebConv K=2, sym-norm, lambda_max=2) on gfx1250.
//   - Sparse propagation: gather(row) * norm_w -> atomicAdd scatter(col).
//     Working set (~70MB) fits the 192MB L2 -> L2-atomic bound, not HBM.
//   - Dense gate GEMMs use V_WMMA_F32_16X16X4_F32 (fp32 WMMA): one wave per
//     16-node tile; gates z and r packed into the 16 N-columns of a single
//     accumulator (cols 0-7 = z, cols 8-15 = r).
//   - Activations use hardware v_rcp_f32 / v_exp_f32 (no OCML div/tanh).
// ---------------------------------------------------------------------------

typedef float v2f __attribute__((ext_vector_type(2)));
typedef float v8f __attribute__((ext_vector_type(8)));

static __device__ __forceinline__ v8f wmma4(v2f a, v2f b, v8f c) {
    // D = A(16x4 f32) * B(4x16 f32) + C(16x16 f32)
    return __builtin_amdgcn_wmma_f32_16x16x4_f32(
        /*neg_a=*/false, a, /*neg_b=*/false, b,
        /*c_mod=*/(short)0, c, /*reuse_a=*/false, /*reuse_b=*/false);
}

static __device__ __forceinline__ float fast_rcp(float x) {
#if __has_builtin(__builtin_amdgcn_rcpf)
    return __builtin_amdgcn_rcpf(x);        // v_rcp_f32
#else
    return 1.0f / x;
#endif
}

static __device__ __forceinline__ float fast_sigmoid(float x) {
    return fast_rcp(1.0f + __expf(-x));
}

static __device__ __forceinline__ float fast_tanh(float x) {
    // tanh(x) = 1 - 2/(e^{2x}+1); saturates correctly: exp->inf => 1, exp->0 => -1
    return 1.0f - 2.0f * fast_rcp(1.0f + __expf(2.0f * x));
}

// ----------------------------- utility kernels ------------------------------

__global__ void k_fill(float* __restrict__ p, float v, int n) {
    int i = blockIdx.x * blockDim.x + threadIdx.x;
    if (i < n) p[i] = v;
}

__global__ void k_degree(const int* __restrict__ ei, float* __restrict__ deg, int E) {
    int e = blockIdx.x * blockDim.x + threadIdx.x;
    if (e < E) atomicAdd(&deg[ei[e]], 1.0f);   // segment_sum(ones, row)
}

__global__ void k_dinv(float* __restrict__ deg, int n) {
    int i = blockIdx.x * blockDim.x + threadIdx.x;
    if (i < n) {
        float d = deg[i];
        deg[i] = (d > 0.0f) ? rsqrtf(d) : 0.0f;   // in-place deg -> d^{-1/2}
    }
}

__global__ void k_normw(const int* __restrict__ ei, const float* __restrict__ dinv,
                        float* __restrict__ nw, int E) {
    int e = blockIdx.x * blockDim.x + threadIdx.x;
    if (e < E) {
        int r = ei[e];
        int c = ei[E + e];
        nw[e] = -(dinv[r] * dinv[c]);
    }
}

// out[col] += w * x[row]   (3 channels)
__global__ void k_scatter3(const int* __restrict__ ei, const float* __restrict__ nw,
                           const float* __restrict__ x, float* __restrict__ out, int E) {
    int e = blockIdx.x * blockDim.x + threadIdx.x;
    if (e >= E) return;
    int r = ei[e], c = ei[E + e];
    float w = nw[e];
    const float* xr = x + (size_t)r * 3;
    float* o = out + (size_t)c * 3;
    atomicAdd(o + 0, w * xr[0]);
    atomicAdd(o + 1, w * xr[1]);
    atomicAdd(o + 2, w * xr[2]);
}

// out[col] += w * h[row]   (8 channels, 2x float4 gather)
__global__ void k_scatter8(const int* __restrict__ ei, const float* __restrict__ nw,
                           const float* __restrict__ h, float* __restrict__ out, int E) {
    int e = blockIdx.x * blockDim.x + threadIdx.x;
    if (e >= E) return;
    int r = ei[e], c = ei[E + e];
    float w = nw[e];
    const float4* hp = reinterpret_cast<const float4*>(h + (size_t)r * 8);
    float4 a = hp[0], b = hp[1];
    float* o = out + (size_t)c * 8;
    atomicAdd(o + 0, w * a.x); atomicAdd(o + 1, w * a.y);
    atomicAdd(o + 2, w * a.z); atomicAdd(o + 3, w * a.w);
    atomicAdd(o + 4, w * b.x); atomicAdd(o + 5, w * b.y);
    atomicAdd(o + 6, w * b.z); atomicAdd(o + 7, w * b.w);
}

// --------------------------- WMMA gate kernels ------------------------------
// Fragment layouts per CDNA5 ISA 7.12.2 (f32, wave32):
//   A 16x4 : lanes 0-15 M=lane, {K0,K1}; lanes 16-31 M=lane-16, {K2,K3}
//   B 4x16 : lanes 0-15 N=lane, rows {K0,K1}; lanes 16-31 N=lane-16, {K2,K3}
//   C/D    : VGPR v holds row v (lanes 0-15) / row v+8 (lanes 16-31), col=lane&15

// Computes Z = sigmoid(cheb_z) (cols 0-7) and HR = H * sigmoid(cheb_r)
// (cols 8-15) for a 16-node tile per wave.
__global__ void k_gate_zr(const float* __restrict__ x,  const float* __restrict__ LX,
                          const float* __restrict__ H,  const float* __restrict__ LH,
                          const float* __restrict__ Wx, const float* __restrict__ Wh,
                          const float* __restrict__ bx, const float* __restrict__ bh,
                          float* __restrict__ Z, float* __restrict__ HR, int nTiles) {
    const int lane = threadIdx.x & 31;
    const int tile = (blockIdx.x * blockDim.x + threadIdx.x) >> 5;
    if (tile >= nTiles) return;                 // wave-uniform: EXEC stays all-1s
    const int half = lane >> 4;                 // 0: K0/K1, rows 0-7 | 1: K2/K3, rows 8-15
    const int m    = lane & 15;
    const int k0   = half * 2;
    const int node = tile * 16 + m;

    // ---- A fragments (per-node rows) ----
    v2f Ax;  Ax.x  = x [node * 3 + k0]; Ax.y  = half ? 0.0f : x [node * 3 + 1]; // ch 3->4 pad
    v2f ALx; ALx.x = LX[node * 3 + k0]; ALx.y = half ? 0.0f : LX[node * 3 + 1];
    const float* hr  = H  + (size_t)node * 8;
    const float* lhr = LH + (size_t)node * 8;
    v2f AH0;  AH0.x  = hr [k0];     AH0.y  = hr [k0 + 1];
    v2f AH1;  AH1.x  = hr [4 + k0]; AH1.y  = hr [4 + k0 + 1];
    v2f ALH0; ALH0.x = lhr[k0];     ALH0.y = lhr[k0 + 1];
    v2f ALH1; ALH1.x = lhr[4 + k0]; ALH1.y = lhr[4 + k0 + 1];

    // ---- B fragments: gate g = 0 (z) for cols 0-7, g = 1 (r) for cols 8-15
    const int g = m >> 3;
    const int h = m & 7;
    const float* wx0 = Wx + (size_t)((g * 2 + 0) * 3) * 8;   // Wx[g][0]  [3,8]
    const float* wx1 = Wx + (size_t)((g * 2 + 1) * 3) * 8;   // Wx[g][1]
    const float* wh0 = Wh + (size_t)((g * 2 + 0) * 8) * 8;   // Wh[g][0]  [8,8]
    const float* wh1 = Wh + (size_t)((g * 2 + 1) * 8) * 8;   // Wh[g][1]
    v2f Bx0;  Bx0.x  = wx0[k0 * 8 + h];       Bx0.y  = half ? 0.0f : wx0[8 + h];
    v2f Bx1;  Bx1.x  = wx1[k0 * 8 + h];       Bx1.y  = half ? 0.0f : wx1[8 + h];
    v2f Bh00; Bh00.x = wh0[k0 * 8 + h];       Bh00.y = wh0[(k0 + 1) * 8 + h];
    v2f Bh01; Bh01.x = wh0[(4 + k0) * 8 + h]; Bh01.y = wh0[(4 + k0 + 1) * 8 + h];
    v2f Bh10; Bh10.x = wh1[k0 * 8 + h];       Bh10.y = wh1[(k0 + 1) * 8 + h];
    v2f Bh11; Bh11.x = wh1[(4 + k0) * 8 + h]; Bh11.y = wh1[(4 + k0 + 1) * 8 + h];

    v8f acc = {};
    acc = wmma4(Ax,   Bx0,  acc);   // x_t   @ Wx[g][0]
    acc = wmma4(ALx,  Bx1,  acc);   // L^x_t @ Wx[g][1]
    acc = wmma4(AH0,  Bh00, acc);   // H     @ Wh[g][0]  (K 0-3)
    acc = wmma4(AH1,  Bh01, acc);   //                   (K 4-7)
    acc = wmma4(ALH0, Bh10, acc);   // L^H   @ Wh[g][1]  (K 0-3)
    acc = wmma4(ALH1, Bh11, acc);   //                   (K 4-7)

    const float bias = bx[g * 8 + h] + bh[g * 8 + h];
#pragma unroll
    for (int v = 0; v < 8; ++v) {
        const int dn = tile * 16 + v + half * 8;    // D row -> node
        const float s = fast_sigmoid(acc[v] + bias);
        if (g == 0) Z [(size_t)dn * 8 + h] = s;
        else        HR[(size_t)dn * 8 + h] = s * H[(size_t)dn * 8 + h];
    }
}

// H_tilde = tanh(cheb(x) + cheb(H*R)); H = Z*H + (1-Z)*H_tilde. Cols 8-15 idle.
__global__ void k_gate_h(const float* __restrict__ x,    const float* __restrict__ LX,
                         const float* __restrict__ HRin, const float* __restrict__ LHR,
                         const float* __restrict__ Wx,   const float* __restrict__ Wh,
                         const float* __restrict__ bx,   const float* __restrict__ bh,
                         const float* __restrict__ Z,    float* __restrict__ H, int nTiles) {
    const int lane = threadIdx.x & 31;
    const int tile = (blockIdx.x * blockDim.x + threadIdx.x) >> 5;
    if (tile >= nTiles) return;                 // wave-uniform
    const int half = lane >> 4;
    const int m    = lane & 15;
    const int k0   = half * 2;
    const int node = tile * 16 + m;

    v2f Ax;  Ax.x  = x [node * 3 + k0]; Ax.y  = half ? 0.0f : x [node * 3 + 1];
    v2f ALx; ALx.x = LX[node * 3 + k0]; ALx.y = half ? 0.0f : LX[node * 3 + 1];
    const float* hr  = HRin + (size_t)node * 8;
    const float* lhr = LHR  + (size_t)node * 8;
    v2f AH0;  AH0.x  = hr [k0];     AH0.y  = hr [k0 + 1];
    v2f AH1;  AH1.x  = hr [4 + k0]; AH1.y  = hr [4 + k0 + 1];
    v2f ALH0; ALH0.x = lhr[k0];     ALH0.y = lhr[k0 + 1];
    v2f ALH1; ALH1.x = lhr[4 + k0]; ALH1.y = lhr[4 + k0 + 1];

    const int   h  = m & 7;
    const float en = (m < 8) ? 1.0f : 0.0f;     // zero B for unused cols 8-15
    const float* wx0 = Wx + (size_t)((2 * 2 + 0) * 3) * 8;   // Wx[2][0]
    const float* wx1 = Wx + (size_t)((2 * 2 + 1) * 3) * 8;
    const float* wh0 = Wh + (size_t)((2 * 2 + 0) * 8) * 8;   // Wh[2][0]
    const float* wh1 = Wh + (size_t)((2 * 2 + 1) * 8) * 8;
    v2f Bx0;  Bx0.x  = en * wx0[k0 * 8 + h];       Bx0.y  = half ? 0.0f : en * wx0[8 + h];
    v2f Bx1;  Bx1.x  = en * wx1[k0 * 8 + h];       Bx1.y  = half ? 0.0f : en * wx1[8 + h];
    v2f Bh00; Bh00.x = en * wh0[k0 * 8 + h];       Bh00.y = en * wh0[(k0 + 1) * 8 + h];
    v2f Bh01; Bh01.x = en * wh0[(4 + k0) * 8 + h]; Bh01.y = en * wh0[(4 + k0 + 1) * 8 + h];
    v2f Bh10; Bh10.x = en * wh1[k0 * 8 + h];       Bh10.y = en * wh1[(k0 + 1) * 8 + h];
    v2f Bh11; Bh11.x = en * wh1[(4 + k0) * 8 + h]; Bh11.y = en * wh1[(4 + k0 + 1) * 8 + h];

    v8f acc = {};
    acc = wmma4(Ax,   Bx0,  acc);
    acc = wmma4(ALx,  Bx1,  acc);
    acc = wmma4(AH0,  Bh00, acc);
    acc = wmma4(AH1,  Bh01, acc);
    acc = wmma4(ALH0, Bh10, acc);
    acc = wmma4(ALH1, Bh11, acc);

    if (m < 8) {                                 // divergence after WMMA is fine
        const float bias = bx[16 + h] + bh[16 + h];
#pragma unroll
        for (int v = 0; v < 8; ++v) {
            const int dn = tile * 16 + v + half * 8;
            const float ht   = fast_tanh(acc[v] + bias);
            const float z    = Z[(size_t)dn * 8 + h];
            const float hold = H[(size_t)dn * 8 + h];
            H[(size_t)dn * 8 + h] = z * hold + (1.0f - z) * ht;   // in-place, no aliasing
        }
    }
}

// ------------------------------ final reduce --------------------------------

__global__ void k_out_init(float* __restrict__ out, const float* __restrict__ lb) {
    if (blockIdx.x == 0 && threadIdx.x == 0) out[0] = lb[0];
}

__global__ void k_reduce(const float* __restrict__ H, const float* __restrict__ lw,
                         float* __restrict__ out, int n) {
    float w0 = lw[0], w1 = lw[1], w2 = lw[2], w3 = lw[3];
    float w4 = lw[4], w5 = lw[5], w6 = lw[6], w7 = lw[7];
    float s = 0.0f;
    for (int i = blockIdx.x * blockDim.x + threadIdx.x; i < n;
         i += gridDim.x * blockDim.x) {
        const float4* hp = reinterpret_cast<const float4*>(H + (size_t)i * 8);
        float4 a = hp[0], b = hp[1];
        s += a.x * w0 + a.y * w1 + a.z * w2 + a.w * w3 +
             b.x * w4 + b.y * w5 + b.z * w6 + b.w * w7;
    }
    // wave32 reduction
    for (int off = 16; off > 0; off >>= 1) s += __shfl_down(s, off);
    if ((threadIdx.x & 31) == 0) atomicAdd(out, s / (float)n);
}

// ------------------------------ host launcher -------------------------------

static inline size_t alignup(size_t x) { return (x + 255) & ~(size_t)255; }

extern "C" void kernel_launch(void* const* d_in, const int* in_sizes, int n_in,
                              void* d_out, int out_size, void* d_ws, size_t ws_size,
                              hipStream_t stream) {
    const float* X  = (const float*)d_in[0];   // [T, N, 3]
    const int*   EI = (const int*)  d_in[1];   // [2, E]
    const float* Wx = (const float*)d_in[2];   // [3,2,3,8]
    const float* Wh = (const float*)d_in[3];   // [3,2,8,8]
    const float* bx = (const float*)d_in[4];   // [3,8]
    const float* bh = (const float*)d_in[5];   // [3,8]
    const float* lw = (const float*)d_in[6];   // [8]
    const float* lb = (const float*)d_in[7];   // scalar
    float* out = (float*)d_out;

    const int T = 8;
    const int N = in_sizes[0] / (T * 3);
    const int E = in_sizes[1] / 2;

    // workspace carve-up (floats), ~24 MB total
    float* ws = (float*)d_ws;
    size_t o = 0;
    float* deg = ws + o; o = alignup(o + (size_t)N);        // deg -> dinv in place
    float* nw  = ws + o; o = alignup(o + (size_t)E);
    float* H   = ws + o; o = alignup(o + (size_t)N * 8);
    float* LX  = ws + o; o = alignup(o + (size_t)N * 3);
    float* LH  = ws + o; o = alignup(o + (size_t)N * 8);
    float* LHR = ws + o; o = alignup(o + (size_t)N * 8);
    float* Zb  = ws + o; o = alignup(o + (size_t)N * 8);
    float* HRb = ws + o; o = alignup(o + (size_t)N * 8);
    (void)ws_size;

    const int BLK = 256;
    auto g1 = [&](int n) { return (n + BLK - 1) / BLK; };

    // ---- one-time graph normalization ----
    k_fill  <<<g1(N),     BLK, 0, stream>>>(deg, 0.0f, N);
    k_fill  <<<g1(N * 8), BLK, 0, stream>>>(H, 0.0f, N * 8);
    k_degree<<<g1(E),     BLK, 0, stream>>>(EI, deg, E);
    k_dinv  <<<g1(N),     BLK, 0, stream>>>(deg, N);
    k_normw <<<g1(E),     BLK, 0, stream>>>(EI, deg, nw, E);

    const int nTiles  = N / 16;                 // N = 100000 is a multiple of 16
    const int gateGrd = (nTiles * 32 + BLK - 1) / BLK;

    for (int t = 0; t < T; ++t) {
        const float* xt = X + (size_t)t * N * 3;
        // L^ x_t  and  L^ H
        k_fill    <<<g1(N * 3), BLK, 0, stream>>>(LX, 0.0f, N * 3);
        k_fill    <<<g1(N * 8), BLK, 0, stream>>>(LH, 0.0f, N * 8);
        k_scatter3<<<g1(E),     BLK, 0, stream>>>(EI, nw, xt, LX, E);
        k_scatter8<<<g1(E),     BLK, 0, stream>>>(EI, nw, H, LH, E);
        // gates z, r (WMMA): Zb, HRb = R (.) H
        k_gate_zr <<<gateGrd,   BLK, 0, stream>>>(xt, LX, H, LH, Wx, Wh, bx, bh,
                                                  Zb, HRb, nTiles);
        // L^ (R (.) H)
        k_fill    <<<g1(N * 8), BLK, 0, stream>>>(LHR, 0.0f, N * 8);
        k_scatter8<<<g1(E),     BLK, 0, stream>>>(EI, nw, HRb, LHR, E);
        // h~ gate + GRU update (WMMA), H updated in place
        k_gate_h  <<<gateGrd,   BLK, 0, stream>>>(xt, LX, HRb, LHR, Wx, Wh, bx, bh,
                                                  Zb, H, nTiles);
    }

    // logit = mean_n(H) . lin_w + lin_b
    k_out_init<<<1,   1,   0, stream>>>(out, lb);
    k_reduce  <<<512, BLK, 0, stream>>>(H, lw, out, N);
}